// EmbedMatcher_82557861363990
// MI455X (gfx1250) — compile-verified
//
#include <hip/hip_runtime.h>
#include <hip/hip_bf16.h>
#include <math.h>

// ---------------------------------------------------------------------------
// EmbedMatcher for MI455X (gfx1250, wave32, WMMA, TDM).
// 4 entities per block: every W B-fragment load feeds 4 WMMAs (4x less L2
// traffic). gate_w1 fragments staged to LDS via TENSOR_LOAD_TO_LDS.
// GEMM1: nbr[BK,128] = cat(rel,ent)[BK,256] @ W^T[256,128]  (f32 WMMA 16x16x4)
// GEMM2: h[BK,64]    = nbr[BK,128] @ w1^T[128,64]           (f32 WMMA 16x16x4)
// ---------------------------------------------------------------------------

#define D       128
#define TWO_D   256
#define KN      50
#define KPAD    64        // 4 M-tiles of 16
#define PAD_IDX 100000
#define ROWSTR  130       // padded LDS row stride (bank-conflict-free)
#define EB      4         // entities per block

typedef __attribute__((ext_vector_type(2))) float v2f;
typedef __attribute__((ext_vector_type(8))) float v8f;
typedef __attribute__((ext_vector_type(4))) unsigned int u32x4;
typedef __attribute__((ext_vector_type(8))) int i32x8;
typedef __attribute__((ext_vector_type(4))) int i32x4;

#if defined(__has_builtin)
#  if __has_builtin(__builtin_amdgcn_tensor_load_to_lds) && \
      __has_builtin(__builtin_amdgcn_s_wait_tensorcnt)
#    define USE_TDM 1
#  endif
#endif
#ifndef USE_TDM
#  define USE_TDM 0
#endif

__device__ __forceinline__ v8f wmma_f32x4(v2f a, v2f b, v8f c) {
  // 8 args: (neg_a, A, neg_b, B, c_mod, C, reuse_a, reuse_b)
  return __builtin_amdgcn_wmma_f32_16x16x4_f32(
      false, a, false, b, (short)0, c, false, false);
}

// ---- Pre-swizzle W (D x 2D, row-major) into per-lane B fragments -----------
// B1[k=f][n=d] = W[d*256+f].  Fragment (ks, nt, lane):
//   n = nt*16 + lane%16 ; f0 = ks*4 + (lane<16 ? 0 : 2)
__global__ void swizzle_W_kernel(const float* __restrict__ W,
                                 v2f* __restrict__ wfrag) {
  int tid = blockIdx.x * blockDim.x + threadIdx.x;
  if (tid >= 64 * 8 * 32) return;
  int lane = tid & 31;
  int nt   = (tid >> 5) & 7;
  int ks   = tid >> 8;
  int n  = nt * 16 + (lane & 15);
  int f0 = ks * 4 + ((lane & 16) ? 2 : 0);
  v2f v;
  v[0] = W[n * TWO_D + f0];
  v[1] = W[n * TWO_D + f0 + 1];
  wfrag[tid] = v;
}

// ---- Pre-swizzle gate_w1 (64 x 128): B2[k=d][n=h] = w1[h*128+d] ------------
__global__ void swizzle_w1_kernel(const float* __restrict__ w1,
                                  v2f* __restrict__ w1frag) {
  int tid = blockIdx.x * blockDim.x + threadIdx.x;
  if (tid >= 32 * 4 * 32) return;
  int lane = tid & 31;
  int nt   = (tid >> 5) & 3;
  int ks   = tid >> 7;
  int n  = nt * 16 + (lane & 15);
  int f0 = ks * 4 + ((lane & 16) ? 2 : 0);
  v2f v;
  v[0] = w1[n * D + f0];
  v[1] = w1[n * D + f0 + 1];
  w1frag[tid] = v;
}

// ---------------------------------------------------------------------------
__global__ __launch_bounds__(128) void embed_matcher_kernel(
    const int*   __restrict__ entity,
    const int*   __restrict__ connections,   // [B, KN, 2]
    const float* __restrict__ emb,           // [100001, 128]
    const float* __restrict__ gcn_w_b,       // [128]
    const float* __restrict__ gcn_b,         // [128]
    const float* __restrict__ gate_b1,       // [64]
    const float* __restrict__ gate_w2,       // [64]
    const float* __restrict__ gate_b2,       // [1]
    const float* __restrict__ gate_temp,     // [1]
    const v2f*   __restrict__ wfrag,         // [64*8*32] swizzled W^T frags
    const v2f*   __restrict__ w1frag,        // [32*4*32] swizzled w1^T frags
    float*       __restrict__ out,           // [B, 128]
    int B)
{
  __shared__ float snbr[KPAD * ROWSTR];      // nbr rows, reused per entity
  __shared__ v2f   sW1[32 * 4 * 32];         // 32KB staged w1 fragments (TDM)
  __shared__ float sself[EB * D];
  __shared__ float sbias[D];
  __shared__ float sscore[KPAD];
  __shared__ float slogit[KPAD];
  __shared__ float swgt[KPAD];
  __shared__ float stmp[KPAD];

  const int tid  = threadIdx.x;
  const int w    = tid >> 5;                 // wave id -> M-tile
  const int lane = tid & 31;
  const int l16  = lane & 15;
  const int hi   = (lane >= 16) ? 1 : 0;
  const int kofs = hi ? 2 : 0;
  const int b0   = blockIdx.x * EB;

  // ---------------- Phase 1: stage small vectors, gather pointers ----------
#if USE_TDM
  if (w == 0) {
    // D# for a 1D 32KB copy: data_size=4B, tensor_dim0=tile_dim0=8192 elems.
    u32x4 g0;
    i32x8 g1;
    i32x4 g2 = (i32x4)0, g3 = (i32x4)0;
    i32x8 g4 = (i32x8)0;
    const unsigned lds_base = (unsigned)(size_t)(void*)&sW1[0];
    const unsigned long long ga = (unsigned long long)(size_t)w1frag;
    const unsigned N = 32u * 4u * 32u * 2u;  // 8192 f32 elements
    g0[0] = 1u;                                        // count=1 (valid)
    g0[1] = lds_base;                                  // lds_addr
    g0[2] = (unsigned)(ga & 0xFFFFFFFFu);              // global_addr lo
    g0[3] = (unsigned)((ga >> 32) & 0x01FFFFFFull) | (2u << 30);  // hi | type=2
    g1[0] = (int)(2u << 16);                           // data_size=4B
    g1[1] = (int)((N & 0xFFFFu) << 16);                // tensor_dim0[15:0]
    g1[2] = (int)((N >> 16) | (1u << 16));             // dim0 hi | tensor_dim1=1
    g1[3] = (int)(N << 16);                            // tile_dim0=8192
    g1[4] = 0;                                         // tile_dim1/2 unused
    g1[5] = (int)N;                                    // tensor_dim0_stride
    g1[6] = 0;
    g1[7] = 0;
    __builtin_amdgcn_tensor_load_to_lds(g0, g1, g2, g3, g4, 0);
  }
#else
  for (int i = tid; i < 32 * 4 * 32; i += 128) sW1[i] = w1frag[i];
#endif

  if (tid < D) sbias[tid] = gcn_w_b[tid];
  int bs[EB];
#pragma unroll
  for (int e = 0; e < EB; ++e) {
    int bb = b0 + e;
    bs[e] = (bb < B) ? bb : (B - 1);
    if (tid < D) sself[e * D + tid] = emb[(size_t)entity[bs[e]] * D + tid];
  }

  // Per-lane gather pointers: wave w owns neighbor rows w*16 .. w*16+15.
  const int r = (w << 4) + l16;
  const float* relp[EB];
  const float* entp[EB];
#pragma unroll
  for (int e = 0; e < EB; ++e) {
    int ridx = PAD_IDX, eidx = PAD_IDX;
    if (r < KN) {
      const int* cp = connections + ((size_t)bs[e] * KN + r) * 2;
      ridx = cp[0];
      eidx = cp[1];
    }
    relp[e] = emb + (size_t)ridx * D;
    entp[e] = emb + (size_t)eidx * D;
  }
  __syncthreads();

  // ---------------- Phase 2: batched GEMM1 (4 entities share B frags) ------
  v8f acc[EB][8];
#pragma unroll
  for (int e = 0; e < EB; ++e)
#pragma unroll
    for (int nt = 0; nt < 8; ++nt) acc[e][nt] = (v8f)(0.0f);

  // ks 0..31 -> rel half (f < 128); ks 32..63 -> ent half. No selects needed.
  for (int ks = 0; ks < 32; ++ks) {
    const int f0 = (ks << 2) + kofs;
    v2f a[EB];
#pragma unroll
    for (int e = 0; e < EB; ++e) a[e] = *(const v2f*)(relp[e] + f0);
    const v2f* bf = wfrag + ((ks << 3) << 5) + lane;
#pragma unroll
    for (int nt = 0; nt < 8; ++nt) {
      v2f bb = bf[nt << 5];
#pragma unroll
      for (int e = 0; e < EB; ++e) acc[e][nt] = wmma_f32x4(a[e], bb, acc[e][nt]);
    }
  }
  for (int ks = 32; ks < 64; ++ks) {
    const int f0 = ((ks - 32) << 2) + kofs;
    v2f a[EB];
#pragma unroll
    for (int e = 0; e < EB; ++e) a[e] = *(const v2f*)(entp[e] + f0);
    const v2f* bf = wfrag + ((ks << 3) << 5) + lane;
#pragma unroll
    for (int nt = 0; nt < 8; ++nt) {
      v2f bb = bf[nt << 5];
#pragma unroll
      for (int e = 0; e < EB; ++e) acc[e][nt] = wmma_f32x4(a[e], bb, acc[e][nt]);
    }
  }

#if USE_TDM
  if (w == 0) __builtin_amdgcn_s_wait_tensorcnt(0);  // sW1 resident before GEMM2
#endif

  // ---------------- Phase 3..5: per-entity epilogue (reuses snbr) ----------
  const float rs = 0.08838834764831845f;  // 1/sqrt(128)
  for (int e = 0; e < EB; ++e) {
    __syncthreads();   // prev entity done with snbr; TDM data visible
    if (tid < KPAD) {
      sscore[tid] = 0.0f;
      slogit[tid] = 0.0f;
    }
    __syncthreads();

    // nbr = acc + gcn_w_b -> LDS; attention-score partials via ds_add_f32.
#pragma unroll
    for (int nt = 0; nt < 8; ++nt) {
      const int col = (nt << 4) + l16;
      const float bb = sbias[col];
      const float sv = sself[e * D + col] * rs;
#pragma unroll
      for (int j = 0; j < 8; ++j) {
        const int row = (w << 4) + j + (hi ? 8 : 0);   // C layout: M=j / j+8
        const float v = acc[e][nt][j] + bb;
        snbr[row * ROWSTR + col] = v;
        atomicAdd(&sscore[row], v * sv);
      }
    }
    __syncthreads();

    // GEMM2: h = nbr @ w1^T  (A frags from padded snbr, B frags from LDS sW1)
    v8f h[4];
#pragma unroll
    for (int nt = 0; nt < 4; ++nt) h[nt] = (v8f)(0.0f);
    const int arow = (w << 4) + l16;
    for (int ks = 0; ks < 32; ++ks) {
      const int f0 = (ks << 2) + kofs;
      v2f a = *(const v2f*)&snbr[arow * ROWSTR + f0];  // ds_load_b64
      const v2f* bf = &sW1[((ks << 2) << 5) + lane];
#pragma unroll
      for (int nt = 0; nt < 4; ++nt) h[nt] = wmma_f32x4(a, bf[nt << 5], h[nt]);
    }

    // relu(h + b1) . w2 -> logits
#pragma unroll
    for (int nt = 0; nt < 4; ++nt) {
      const int col = (nt << 4) + l16;
      const float b1v = gate_b1[col];
      const float w2v = gate_w2[col];
#pragma unroll
      for (int j = 0; j < 8; ++j) {
        const int row = (w << 4) + j + (hi ? 8 : 0);
        const float hv = fmaxf(h[nt][j] + b1v, 0.0f);
        atomicAdd(&slogit[row], hv * w2v);
      }
    }
    __syncthreads();

    // softmax over real K + sigmoid gate -> combined weight
    if (tid == 0) {
      float mx = -3.0e38f;
      for (int i = 0; i < KN; ++i) {
        stmp[i] = sscore[i];
        mx = fmaxf(mx, stmp[i]);
      }
      float sum = 0.0f;
      for (int i = 0; i < KN; ++i) {
        float ev = expf(stmp[i] - mx);
        stmp[i] = ev;
        sum += ev;
      }
      const float inv  = 1.0f / sum;
      const float b2v  = gate_b2[0];
      const float itmp = 1.0f / gate_temp[0];
      for (int i = 0; i < KN; ++i) {
        const float g = 1.0f / (1.0f + expf(-(slogit[i] + b2v) * itmp));
        swgt[i] = stmp[i] * inv * g;
      }
    }
    __syncthreads();

    // masked max over real neighbors, tanh, store
    if (b0 + e < B) {
      const int d = tid;
      float m = -3.0e38f;
      for (int i = 0; i < KN; ++i) {
        m = fmaxf(m, swgt[i] * snbr[i * ROWSTR + d]);
      }
      out[(size_t)(b0 + e) * D + d] = tanhf(m + gcn_b[d]);
    }
  }
}

// ---------------------------------------------------------------------------
extern "C" void kernel_launch(void* const* d_in, const int* in_sizes, int n_in,
                              void* d_out, int out_size, void* d_ws, size_t ws_size,
                              hipStream_t stream) {
  const int*   entity      = (const int*)  d_in[0];
  const int*   connections = (const int*)  d_in[1];
  const float* emb         = (const float*)d_in[2];
  const float* W           = (const float*)d_in[3];   // gcn_w_W [128,256]
  const float* gcn_w_b     = (const float*)d_in[4];
  const float* gcn_b       = (const float*)d_in[5];
  const float* gate_w1     = (const float*)d_in[6];   // [64,128]
  const float* gate_b1     = (const float*)d_in[7];
  const float* gate_w2     = (const float*)d_in[8];   // [1,64]
  const float* gate_b2     = (const float*)d_in[9];
  const float* gate_temp   = (const float*)d_in[10];
  float*       out         = (float*)d_out;

  const int B = in_sizes[0];

  // Workspace: swizzled WMMA B-fragments (recomputed every call; deterministic)
  v2f* wfrag  = (v2f*)d_ws;                                    // 128KB
  v2f* w1frag = (v2f*)((char*)d_ws + (size_t)64 * 8 * 32 * 8); // +32KB
  (void)ws_size; (void)n_in; (void)out_size;

  swizzle_W_kernel <<<dim3((64 * 8 * 32 + 255) / 256), dim3(256), 0, stream>>>(W, wfrag);
  swizzle_w1_kernel<<<dim3((32 * 4 * 32 + 255) / 256), dim3(256), 0, stream>>>(gate_w1, w1frag);

  embed_matcher_kernel<<<dim3((B + EB - 1) / EB), dim3(128), 0, stream>>>(
      entity, connections, emb, gcn_w_b, gcn_b,
      gate_b1, gate_w2, gate_b2, gate_temp,
      wfrag, w1frag, out, B);
}